// SelfAttention3D_3917010174542
// MI455X (gfx1250) — compile-verified
//
#include <hip/hip_runtime.h>

// SelfAttention3D, B=4 C=256 N=16^3=4096 INTER=32, fp32 end-to-end.
// Stage 1: WMMA_F32_16x16x4 GEMM for q/k/v projections -> d_ws (~21 MB, L2-resident).
// Stage 2: fused flash-attention (online softmax) with fp32 WMMA for Q^T*K and V*P^T.
// attn matrix (268 MB) is never materialized -> HBM traffic ~ x-in + out only.

typedef __attribute__((ext_vector_type(2))) float v2f;
typedef __attribute__((ext_vector_type(8))) float v8f;

#define NB   4      // batch
#define CCH  256    // channels
#define CI   32     // inter (q/k) channels
#define NN   4096   // spatial positions
#define NT   256    // NN / 16 tiles

static __device__ __forceinline__ v8f wmma_f32(v2f a, v2f b, v8f c) {
  // V_WMMA_F32_16X16X4_F32: (neg_a, A, neg_b, B, c_mod, C, reuse_a, reuse_b)
  return __builtin_amdgcn_wmma_f32_16x16x4_f32(false, a, false, b, (short)0, c,
                                               false, false);
}

// ---------------- Stage 1: q/k/v projections ----------------
// grid = B * NT * 20 tiles, 1 wave each.
// mt 0-1 -> q rows, 2-3 -> k rows, 4-19 -> v rows. out[m,n] = sum_c W[m,c]*x[b,c,n] + bias[m]
__global__ __launch_bounds__(32)
void proj_kernel(const float* __restrict__ x,
                 const float* __restrict__ Wq, const float* __restrict__ bq,
                 const float* __restrict__ Wk, const float* __restrict__ bk,
                 const float* __restrict__ Wv, const float* __restrict__ bv,
                 float* __restrict__ ws)
{
  const int lane = threadIdx.x & 31;
  const int ln   = lane & 15;   // n (col) / m (row) sub-index
  const int hl   = lane >> 4;   // half: selects K pair {0,1} vs {2,3}

  const int tile = blockIdx.x;
  const int mt = tile % 20;
  const int nt = (tile / 20) % NT;
  const int b  = tile / (20 * NT);

  const float* W; const float* bias; float* outp; int mbase; int orows;
  if (mt < 2)      { W = Wq; bias = bq; outp = ws;                       mbase = mt * 16;       orows = CI;  }
  else if (mt < 4) { W = Wk; bias = bk; outp = ws + (size_t)NB*CI*NN;    mbase = (mt - 2) * 16; orows = CI;  }
  else             { W = Wv; bias = bv; outp = ws + (size_t)2*NB*CI*NN;  mbase = (mt - 4) * 16; orows = CCH; }

  const int n0 = nt * 16;
  const float* xb = x + (size_t)b * CCH * NN + n0 + ln;     // B operand column
  const float* wp = W + (size_t)(mbase + ln) * CCH;         // A operand row

  v8f acc = {};
#pragma unroll 8
  for (int kb = 0; kb < CCH; kb += 4) {
    const int k0 = kb + 2 * hl;
    v2f a;  a.x  = wp[k0];                 a.y  = wp[k0 + 1];           // contiguous f32x2
    v2f bb; bb.x = xb[(size_t)k0 * NN];    bb.y = xb[(size_t)(k0+1)*NN];
    acc = wmma_f32(a, bb, acc);
  }

  float* op = outp + (size_t)b * orows * NN + n0 + ln;
#pragma unroll
  for (int r = 0; r < 8; ++r) {
    const int m = mbase + r + 8 * hl;
    op[(size_t)m * NN] = acc[r] + bias[m];
  }
}

// ---------------- Stage 2: fused attention ----------------
// grid = B * NT blocks, 128 threads (4 waves). Each wave owns 64 value channels
// (4 accumulator tiles) and redundantly computes the shared 16x16 score tile.
__global__ __launch_bounds__(128)
void attn_kernel(const float* __restrict__ x, const float* __restrict__ ws,
                 const float* __restrict__ gamma, float* __restrict__ outp)
{
  __shared__ float lds_p[4][256];   // per-wave P tile staging (transpose S->B layout)
  __shared__ float lds_row[4][16];  // per-wave per-query broadcast (corr / 1/l)

  const int wv   = threadIdx.x >> 5;
  const int lane = threadIdx.x & 31;
  const int ln   = lane & 15;
  const int hl   = lane >> 4;

  const int it = blockIdx.x % NT;
  const int b  = blockIdx.x / NT;
  const int i0 = it * 16;

  const float* qp = ws + (size_t)b * CI * NN;
  const float* kp = ws + (size_t)NB * CI * NN + (size_t)b * CI * NN;
  const float* vp = ws + (size_t)2 * NB * CI * NN + (size_t)b * CCH * NN;

  const float scale = 0.17677669529663687f;   // 1/sqrt(32), folded into Q

  // Q^T as A-operand chunks: A[m=i][k=c], K=32 -> 8 chunks of K=4
  v2f qa[8];
#pragma unroll
  for (int kk = 0; kk < 8; ++kk) {
    const int c = kk * 4 + 2 * hl;
    qa[kk].x = qp[(size_t)c       * NN + i0 + ln] * scale;
    qa[kk].y = qp[(size_t)(c + 1) * NN + i0 + ln] * scale;
  }

  v8f o[4];
  const v8f vzero = {};
#pragma unroll
  for (int t = 0; t < 4; ++t) o[t] = vzero;

  float mrun[8], lrun[8];
#pragma unroll
  for (int r = 0; r < 8; ++r) { mrun[r] = -1e30f; lrun[r] = 0.0f; }

  for (int jt = 0; jt < NT; ++jt) {
    const int j0 = jt * 16;

    // S = (Q*scale)^T K  -- 8 fp32 WMMAs, rows = queries, cols = keys
    v8f s = vzero;
#pragma unroll
    for (int kk = 0; kk < 8; ++kk) {
      const int c = kk * 4 + 2 * hl;
      v2f kb;
      kb.x = kp[(size_t)c       * NN + j0 + ln];
      kb.y = kp[(size_t)(c + 1) * NN + j0 + ln];
      s = wmma_f32(qa[kk], kb, s);
    }

    // Online softmax per row (query). Row r lives on one 16-lane half.
    float corr[8];
#pragma unroll
    for (int r = 0; r < 8; ++r) {
      float sv = s[r];
      float rm = sv;
      rm = fmaxf(rm, __shfl_xor(rm, 1));
      rm = fmaxf(rm, __shfl_xor(rm, 2));
      rm = fmaxf(rm, __shfl_xor(rm, 4));
      rm = fmaxf(rm, __shfl_xor(rm, 8));
      const float mnew = fmaxf(mrun[r], rm);
      const float cf = __expf(mrun[r] - mnew);
      const float p  = __expf(sv - mnew);
      float rs = p;
      rs += __shfl_xor(rs, 1); rs += __shfl_xor(rs, 2);
      rs += __shfl_xor(rs, 4); rs += __shfl_xor(rs, 8);
      lrun[r] = lrun[r] * cf + rs;
      mrun[r] = mnew;
      corr[r] = cf;
      s[r] = p;   // s now holds P
    }

    // Transpose P through LDS: S layout has query on rows, O needs query on cols.
#pragma unroll
    for (int r = 0; r < 8; ++r) lds_p[wv][(r + 8 * hl) * 16 + ln] = s[r];
    if (ln == 0) {
#pragma unroll
      for (int r = 0; r < 8; ++r) lds_row[wv][r + 8 * hl] = corr[r];
    }
    __syncthreads();

    const float corr_i = lds_row[wv][ln];   // correction for query i = ln
    v2f pb[4];                              // P^T as B-operand: b[k=j][n=i]
#pragma unroll
    for (int kk = 0; kk < 4; ++kk) {
      const int j = kk * 4 + 2 * hl;
      pb[kk].x = lds_p[wv][ln * 16 + j];
      pb[kk].y = lds_p[wv][ln * 16 + j + 1];
    }
    __syncthreads();   // protect lds_p/lds_row before next iteration's writes

    // O[c,i] = O[c,i]*corr_i + V[c,j] P^T[j,i]  -- 16 fp32 WMMAs per wave
    const float* vbb = vp + j0;
#pragma unroll
    for (int t = 0; t < 4; ++t) {
      const int ct = wv * 4 + t;
      o[t] = o[t] * corr_i;
      const float* vrow = vbb + (size_t)(ct * 16 + ln) * NN;
#pragma unroll
      for (int kk = 0; kk < 4; ++kk) {
        const int j = kk * 4 + 2 * hl;
        v2f va; va.x = vrow[j]; va.y = vrow[j + 1];   // contiguous f32x2
        o[t] = wmma_f32(va, pb[kk], o[t]);
      }
    }
  }

  // Normalize by row sums, residual + gamma, store.
  if (ln == 0) {
#pragma unroll
    for (int r = 0; r < 8; ++r) lds_row[wv][r + 8 * hl] = lrun[r];
  }
  __syncthreads();
  const float inv_l = 1.0f / lds_row[wv][ln];
  const float g = gamma[0];

  const float* xb = x    + (size_t)b * CCH * NN + i0 + ln;
  float*       ob = outp + (size_t)b * CCH * NN + i0 + ln;
#pragma unroll
  for (int t = 0; t < 4; ++t) {
#pragma unroll
    for (int r = 0; r < 8; ++r) {
      const int c = (wv * 4 + t) * 16 + r + 8 * hl;
      ob[(size_t)c * NN] = g * (o[t][r] * inv_l) + xb[(size_t)c * NN];
    }
  }
}

extern "C" void kernel_launch(void* const* d_in, const int* in_sizes, int n_in,
                              void* d_out, int out_size, void* d_ws, size_t ws_size,
                              hipStream_t stream) {
  const float* x     = (const float*)d_in[0];
  const float* Wq    = (const float*)d_in[1];
  const float* bq    = (const float*)d_in[2];
  const float* Wk    = (const float*)d_in[3];
  const float* bk    = (const float*)d_in[4];
  const float* Wv    = (const float*)d_in[5];
  const float* bv    = (const float*)d_in[6];
  const float* gamma = (const float*)d_in[7];
  float* outp = (float*)d_out;
  float* ws   = (float*)d_ws;   // needs (2*4*32*4096 + 4*256*4096)*4 B ~= 21 MB

  (void)in_sizes; (void)n_in; (void)out_size; (void)ws_size;

  // Stage 1: q/k/v projections (B * NT * 20 tiles, 1 wave each)
  proj_kernel<<<dim3(NB * NT * 20), dim3(32), 0, stream>>>(x, Wq, bq, Wk, bk, Wv, bv, ws);
  // Stage 2: fused attention (B * NT blocks, 4 waves each)
  attn_kernel<<<dim3(NB * NT), dim3(128), 0, stream>>>(x, ws, gamma, outp);
}